// FOG_52115133170148
// MI455X (gfx1250) — compile-verified
//
#include <hip/hip_runtime.h>
#include <stdint.h>

// ---------------- radix sort configuration ----------------
#define RS_THREADS 256
#define RS_ELEMS   16
#define RS_CHUNK   (RS_THREADS * RS_ELEMS)   // 4096 elements / 16KB per block

#define AS_GLOBAL __attribute__((address_space(1)))
#define AS_LDS    __attribute__((address_space(3)))

typedef int fog_v4i __attribute__((vector_size(16)));   // b128 payload type

// CDNA5 async global->LDS copy (ASYNCcnt-tracked). Guarded so the file still
// compiles if this toolchain does not declare the builtin.
#if defined(__has_builtin)
#if __has_builtin(__builtin_amdgcn_global_load_async_to_lds_b128)
#define FOG_ASYNC_LDS 1
#endif
#endif
#ifndef FOG_ASYNC_LDS
#define FOG_ASYNC_LDS 0
#endif

__device__ __forceinline__ void fog_wait_async0() {
#if FOG_ASYNC_LDS
#if __has_builtin(__builtin_amdgcn_s_wait_asynccnt)
    __builtin_amdgcn_s_wait_asynccnt(0);
#else
    asm volatile("s_wait_asynccnt 0x0" ::: "memory");
#endif
#endif
}

// Stage one RS_CHUNK block of u32 keys into LDS. Full blocks go through the
// async global->LDS path (coalesced b128 per lane); the tail block uses
// guarded direct loads so async never touches out-of-range addresses.
__device__ __forceinline__ void fog_stage_chunk(const uint32_t* __restrict__ src,
                                                uint32_t* stage, int base, int n) {
    int tid = threadIdx.x;
#if FOG_ASYNC_LDS
    if (base + RS_CHUNK <= n) {
#pragma unroll
        for (int e = 0; e < RS_ELEMS / 4; ++e) {
            int idx = (e * RS_THREADS + tid) * 4;     // 16B per lane, coalesced
            __builtin_amdgcn_global_load_async_to_lds_b128(
                (AS_GLOBAL fog_v4i*)(src + base + idx),
                (AS_LDS fog_v4i*)(stage + idx), 0, 0);
        }
        fog_wait_async0();
        return;
    }
#endif
    for (int e = 0; e < RS_ELEMS; ++e) {
        int i = e * RS_THREADS + tid;
        if (base + i < n) stage[i] = src[base + i];
    }
}

// ---------------------------------------------------------------------------
// Per-block 256-bin histogram of one 8-bit digit, input staged through LDS.
// Digit-major layout hist[digit*numBlocks + block] -> one flat exclusive scan
// yields scatter bases. (Defined first so the disasm snippet shows staging.)
// ---------------------------------------------------------------------------
__global__ void fog_hist(const uint32_t* __restrict__ src,
                         uint32_t* __restrict__ hist,
                         int n, int shift, int numBlocks) {
    __shared__ uint32_t h[256];
    __shared__ uint32_t stage[RS_CHUNK];
    int tid = threadIdx.x, blk = blockIdx.x;
    h[tid] = 0u;
    int base = blk * RS_CHUNK;
    fog_stage_chunk(src, stage, base, n);
    __syncthreads();
    int lim = n - base; if (lim > RS_CHUNK) lim = RS_CHUNK;
    for (int e = 0; e < RS_ELEMS; ++e) {
        int i = e * RS_THREADS + tid;
        if (i < lim) {
            uint32_t d = (stage[i] >> shift) & 255u;
            atomicAdd(&h[d], 1u);
        }
    }
    __syncthreads();
    hist[(uint32_t)tid * (uint32_t)numBlocks + (uint32_t)blk] = h[tid];
}

// ---------------------------------------------------------------------------
// Scatter pass: block pulls its 256 scanned bases into LDS, input staged via
// async LDS copy, then each element grabs a slot with an LDS atomic.
// ---------------------------------------------------------------------------
__global__ void fog_scatter(const uint32_t* __restrict__ src,
                            uint32_t* __restrict__ dst,
                            const uint32_t* __restrict__ hist,
                            int n, int shift, int numBlocks) {
    __shared__ uint32_t off[256];
    __shared__ uint32_t stage[RS_CHUNK];
    int tid = threadIdx.x, blk = blockIdx.x;
    off[tid] = hist[(uint32_t)tid * (uint32_t)numBlocks + (uint32_t)blk];
    int base = blk * RS_CHUNK;
    fog_stage_chunk(src, stage, base, n);
    __syncthreads();
    int lim = n - base; if (lim > RS_CHUNK) lim = RS_CHUNK;
    for (int e = 0; e < RS_ELEMS; ++e) {
        int i = e * RS_THREADS + tid;
        if (i < lim) {
            uint32_t v = stage[i];
            uint32_t d = (v >> shift) & 255u;
            uint32_t p = atomicAdd(&off[d], 1u);
            dst[p] = v;
        }
    }
}

// ---------------------------------------------------------------------------
// wave32 shuffle scan + two-level block scan (warpSize == 32 on gfx1250)
// ---------------------------------------------------------------------------
__device__ __forceinline__ uint32_t fog_wave_incl_scan(uint32_t v) {
    const int lane = (int)(threadIdx.x & 31u);
#pragma unroll
    for (int d = 1; d < 32; d <<= 1) {
        uint32_t t = __shfl_up(v, (unsigned)d, 32);
        if (lane >= d) v += t;
    }
    return v;
}

template <int NT>
__device__ __forceinline__ uint32_t fog_block_incl_scan(uint32_t v, uint32_t* wsum) {
    const int lane = (int)(threadIdx.x & 31u);
    const int wid  = (int)(threadIdx.x >> 5);
    uint32_t incl = fog_wave_incl_scan(v);
    if (lane == 31) wsum[wid] = incl;
    __syncthreads();
    if (wid == 0) {
        uint32_t w = (lane < (NT / 32)) ? wsum[lane] : 0u;
        w = fog_wave_incl_scan(w);
        if (lane < (NT / 32)) wsum[lane] = w;
    }
    __syncthreads();
    return incl + ((wid > 0) ? wsum[wid - 1] : 0u);
}

// ---------------------------------------------------------------------------
// Single-workgroup in-place exclusive scan: 1024 threads x 4 elements per
// iteration (4096/iter), thread-sequential + block shuffle scan + carry.
// ---------------------------------------------------------------------------
__global__ void fog_scan1block(uint32_t* __restrict__ data, int total) {
    const int T = 1024, E = 4;
    __shared__ uint32_t wsum[T / 32];
    __shared__ uint32_t carryS;
    int tid = threadIdx.x;
    if (tid == 0) carryS = 0u;
    __syncthreads();
    for (int base = 0; base < total; base += T * E) {
        int i0 = base + tid * E;
        uint32_t v[E];
        uint32_t s = 0u;
#pragma unroll
        for (int e = 0; e < E; ++e) {
            int i = i0 + e;
            v[e] = (i < total) ? data[i] : 0u;
            s += v[e];
        }
        uint32_t incl = fog_block_incl_scan<T>(s, wsum);
        uint32_t c = carryS;
        uint32_t running = c + incl - s;       // exclusive base for this thread
#pragma unroll
        for (int e = 0; e < E; ++e) {
            int i = i0 + e;
            if (i < total) data[i] = running;
            running += v[e];
        }
        __syncthreads();
        if (tid == T - 1) carryS = c + incl;
        __syncthreads();
    }
}

// ---------------------------------------------------------------------------
// Build packed 32-bit keys:  packed = parent_key << 3 | octant,
// parent_key = ((b*128 + x>>1)*128 + y>>1)*128 + z>>1  < 2^29
// ---------------------------------------------------------------------------
__global__ void fog_build_keys(const int* __restrict__ coords,
                               uint32_t* __restrict__ keys, int n) {
    int i = blockIdx.x * blockDim.x + threadIdx.x;
    if (i >= n) return;
    int4 c = ((const int4*)coords)[i];
    uint32_t b = (uint32_t)c.x, x = (uint32_t)c.y,
             y = (uint32_t)c.z, z = (uint32_t)c.w;
    uint32_t pos = (x & 1u) | ((y & 1u) << 1) | ((z & 1u) << 2);
    uint32_t key = (((b * 128u + (x >> 1)) * 128u + (y >> 1)) * 128u + (z >> 1));
    keys[i] = (key << 3) | pos;
}

// ---------------------------------------------------------------------------
// Segment-boundary flags + block-local inclusive rank scan.
// Thread owns 16 contiguous elements so the sequential scan preserves order.
// ---------------------------------------------------------------------------
__global__ void fog_flag_rank(const uint32_t* __restrict__ keys,
                              uint32_t* __restrict__ ranks,
                              uint32_t* __restrict__ blockSums, int n) {
    __shared__ uint32_t wsum[RS_THREADS / 32];
    int blk = blockIdx.x, tid = threadIdx.x;
    int base = blk * RS_CHUNK + tid * RS_ELEMS;
    uint32_t inclLocal[RS_ELEMS];
    uint32_t local = 0u;
#pragma unroll
    for (int e = 0; e < RS_ELEMS; ++e) {
        int i = base + e;
        uint32_t f = 0u;
        if (i < n) {
            uint32_t k = keys[i] >> 3;
            f = (i == 0) ? 1u : ((k != (keys[i - 1] >> 3)) ? 1u : 0u);
        }
        local += f;
        inclLocal[e] = local;
    }
    uint32_t incl = fog_block_incl_scan<RS_THREADS>(local, wsum);
    uint32_t threadExcl = incl - local;
#pragma unroll
    for (int e = 0; e < RS_ELEMS; ++e) {
        int i = base + e;
        if (i < n) ranks[i] = threadExcl + inclLocal[e];
    }
    if (tid == RS_THREADS - 1) blockSums[blk] = incl;
}

// ---------------------------------------------------------------------------
// Pre-fill output: coords = -1 (jnp.unique fill_value), feats = 0.
// ---------------------------------------------------------------------------
__global__ void fog_init_out(float* __restrict__ outCoords,
                             float* __restrict__ outFeats, int n) {
    int i = blockIdx.x * blockDim.x + threadIdx.x;
    if (i >= n) return;
    float4 m = make_float4(-1.f, -1.f, -1.f, -1.f);
    ((float4*)outCoords)[i] = m;
    outFeats[i] = 0.f;
}

// ---------------------------------------------------------------------------
// Finalize: rank = (block-inclusive rank + scanned block offset) - 1.
// atomicAdd contributions (powers of two -> exact, deterministic); segment
// heads decode & write the parent coords.
// ---------------------------------------------------------------------------
__global__ void fog_finalize(const uint32_t* __restrict__ keys,
                             const uint32_t* __restrict__ ranks,
                             const uint32_t* __restrict__ blockOff,
                             const float* __restrict__ kern,
                             float* __restrict__ outCoords,
                             float* __restrict__ outFeats, int n) {
    int i = blockIdx.x * blockDim.x + threadIdx.x;
    if (i >= n) return;
    uint32_t p   = keys[i];
    uint32_t key = p >> 3;
    uint32_t pos = p & 7u;
    uint32_t r   = ranks[i] + blockOff[i >> 12] - 1u;  // /RS_CHUNK
    float contrib = (float)(1u << pos) * kern[pos];
    atomicAdd(&outFeats[r], contrib);
    bool first = (i == 0) || ((keys[i - 1] >> 3) != key);
    if (first) {
        uint32_t z = key & 127u;
        uint32_t t = key >> 7;
        uint32_t y = t & 127u;  t >>= 7;
        uint32_t x = t & 127u;
        uint32_t b = t >> 7;
        float4 oc = make_float4((float)b, (float)x, (float)y, (float)z);
        ((float4*)outCoords)[r] = oc;
    }
}

// ---------------------------------------------------------------------------
extern "C" void kernel_launch(void* const* d_in, const int* in_sizes, int n_in,
                              void* d_out, int out_size, void* d_ws, size_t ws_size,
                              hipStream_t stream) {
    const int*   coords = (const int*)d_in[0];
    const float* kern   = (const float*)d_in[1];
    int n = in_sizes[0] / 4;

    float* out       = (float*)d_out;
    float* outCoords = out;                          // 4*n floats
    float* outFeats  = out + (size_t)4 * (size_t)n;  // n floats

    int nb = (n + RS_CHUNK - 1) / RS_CHUNK;

    // workspace (u32): keysA[n] keysB[n] ranks[n] hist[nb*256] bsums[nb]
    uint32_t* ws    = (uint32_t*)d_ws;
    uint32_t* keysA = ws;
    uint32_t* keysB = ws + (size_t)n;
    uint32_t* ranks = ws + (size_t)2 * (size_t)n;
    uint32_t* hist  = ws + (size_t)3 * (size_t)n;
    uint32_t* bsums = hist + (size_t)nb * 256u;

    const int tpb = 256;
    int g1 = (n + tpb - 1) / tpb;

    fog_build_keys<<<g1, tpb, 0, stream>>>(coords, keysA, n);

    uint32_t* src = keysA;
    uint32_t* dst = keysB;
    for (int pass = 0; pass < 4; ++pass) {
        int shift = pass * 8;
        fog_hist<<<nb, RS_THREADS, 0, stream>>>(src, hist, n, shift, nb);
        fog_scan1block<<<1, 1024, 0, stream>>>(hist, nb * 256);
        fog_scatter<<<nb, RS_THREADS, 0, stream>>>(src, dst, hist, n, shift, nb);
        uint32_t* t = src; src = dst; dst = t;
    }
    // after 4 passes sorted data is back in keysA (== src)

    fog_flag_rank<<<nb, RS_THREADS, 0, stream>>>(src, ranks, bsums, n);
    fog_scan1block<<<1, 1024, 0, stream>>>(bsums, nb);
    fog_init_out<<<g1, tpb, 0, stream>>>(outCoords, outFeats, n);
    fog_finalize<<<g1, tpb, 0, stream>>>(src, ranks, bsums, kern,
                                         outCoords, outFeats, n);
}